// QMLLayer_79946521248172
// MI455X (gfx1250) — compile-verified
//
#include <hip/hip_runtime.h>
#include <math.h>

typedef float v2f __attribute__((ext_vector_type(2)));
typedef float v8f __attribute__((ext_vector_type(8)));

// ---------------------------------------------------------------------------
// D = A(16x4) * B(4x16) + C(16x16), all f32, wave32 WMMA.
// ---------------------------------------------------------------------------
__device__ inline v8f wmma4(v2f a, v2f b, v8f c) {
    return __builtin_amdgcn_wmma_f32_16x16x4_f32(
        /*neg_a=*/false, a, /*neg_b=*/false, b,
        /*c_mod=*/(short)0, c, /*reuse_a=*/false, /*reuse_b=*/false);
}

// Swap lane halves (lane ^ 16) via ds_swizzle SWAPX16: xor=0x10,or=0,and=0x1f.
__device__ inline float swap16(float v) {
    return __int_as_float(__builtin_amdgcn_ds_swizzle(__float_as_int(v), 0x401F));
}

// ---------------------------------------------------------------------------
// Build the two fixed 16x16 complex unitaries G_l = C3*C2*C1*(⊗_q RZ·RY)
// from weights[2][4][2].  g layout: [l*3 + (0=re,1=im,2=-im)][256] row-major.
// 512 threads, one complex entry each.  Hardware sincos (runs once; angles
// are in [0,pi) so the fast path is plenty accurate).
// ---------------------------------------------------------------------------
__global__ void build_G(const float* __restrict__ w, float* __restrict__ g) {
    int t = threadIdx.x;
    if (t >= 512) return;
    int l = t >> 8;
    int e = t & 255;
    int r = e >> 4;   // row (output component)
    int c = e & 15;   // col (input component)

    // Compose CNOT row-permutations: G[r][c] = (⊗U)[p1(p2(p3(r)))][c]
    int j = r;
    if (j & 2) j ^= 1;  // CNOT(q2->q3): ctrl bit1, tgt bit0
    if (j & 4) j ^= 2;  // CNOT(q1->q2): ctrl bit2, tgt bit1
    if (j & 8) j ^= 4;  // CNOT(q0->q1): ctrl bit3, tgt bit2

    float ar = 1.0f, ai = 0.0f;
    #pragma unroll
    for (int q = 0; q < 4; ++q) {
        float ty = w[(l * 4 + q) * 2 + 0] * 0.5f;
        float tz = w[(l * 4 + q) * 2 + 1] * 0.5f;
        float cy, sy, cz, sz;
        __sincosf(ty, &sy, &cy);
        __sincosf(tz, &sz, &cz);
        int br = (j >> (3 - q)) & 1;
        int bc = (c >> (3 - q)) & 1;
        // U = RZ * RY:  row0 = e^{-i tz}*(bc ? -sy : cy);  row1 = e^{+i tz}*(bc ? cy : sy)
        float ur, ui;
        if (br == 0) { float m = bc ? -sy : cy; ur = m * cz; ui = -m * sz; }
        else         { float m = bc ?  cy : sy; ur = m * cz; ui =  m * sz; }
        float nr = ar * ur - ai * ui;
        float ni = ar * ui + ai * ur;
        ar = nr; ai = ni;
    }
    g[(l * 3 + 0) * 256 + e] = ar;
    g[(l * 3 + 1) * 256 + e] = ai;
    g[(l * 3 + 2) * 256 + e] = -ai;
}

// ---------------------------------------------------------------------------
// Main kernel: each wave simulates 16 samples.
//   M = output state component (16), N = sample (16), K = input component (16).
// C/D layout: lane L holds comps (r + 8*(L>=16)), sample L%16, r=0..7.
// B layout:   chunk k, elem j: lane L holds comp 4k+j+2*(L>=16), sample L%16.
// A layout:   lane L holds row L%16, cols {4k+2h, 4k+2h+1} per chunk.
// ---------------------------------------------------------------------------
__global__ void __launch_bounds__(256) qml_kernel(const float* __restrict__ x,
                                                  const float* __restrict__ g,
                                                  float* __restrict__ out) {
    const int lane = threadIdx.x & 31;
    const int wave = threadIdx.x >> 5;
    const int n    = lane & 15;            // sample within tile
    const bool hi  = lane >= 16;
    const int cb   = hi ? 2 : 0;
    const int samp = blockIdx.x * 128 + wave * 16 + n;

    // ---- per-sample angles (lane n and n+16 redundantly share sample n) ----
    float4 xv = reinterpret_cast<const float4*>(x)[samp];
    float c0, s0, c1, s1, c2, s2, c3, s3;
    __sincosf(xv.x * 0.5f, &s0, &c0);
    __sincosf(xv.y * 0.5f, &s1, &c1);
    __sincosf(xv.z * 0.5f, &s2, &c2);
    __sincosf(xv.w * 0.5f, &s3, &c3);

    // ---- A operands for G0 (re, im, -im pre-negated), 4 K-chunks ----
    v2f Ar[4], Ai[4], An[4];
    #pragma unroll
    for (int k = 0; k < 4; ++k) {
        int off = n * 16 + 4 * k + cb;
        Ar[k] = *reinterpret_cast<const v2f*>(g + 0 * 256 + off);
        Ai[k] = *reinterpret_cast<const v2f*>(g + 1 * 256 + off);
        An[k] = *reinterpret_cast<const v2f*>(g + 2 * 256 + off);
    }

    // ---- initial product state v0 = ⊗RX(x)|0..0>, directly in B layout ----
    // idx = 4k + jj + 2*hi: bit3=(k>>1), bit2=(k&1), bit1=hi, bit0=jj  (disjoint!)
    // amp(idx) = (-i)^popc(idx) * prod of (bit ? s : c); quadrant folds at compile time.
    const float f2  = hi ? s2 : c2;
    const float ab0 = c0 * c1, ab1 = c0 * s1, ab2 = s0 * c1, ab3 = s0 * s1;
    const float ab[4] = {ab0, ab1, ab2, ab3};   // indexed by k (bits 3,2 of idx)
    const float fd0 = f2 * c3, fd1 = f2 * s3;   // indexed by jj (bit 0)
    v2f Br[4], Bi[4];
    #pragma unroll
    for (int k = 0; k < 4; ++k) {
        #pragma unroll
        for (int jj = 0; jj < 2; ++jj) {
            const float m = ab[k] * (jj ? fd1 : fd0);
            const int cpc = ((k >> 1) & 1) + (k & 1) + jj;   // popc without hi bit
            const int ql = cpc & 3;           // quadrant, low half
            const int qh = (cpc + 1) & 3;     // quadrant, high half
            const float rl = (ql == 0) ? m : (ql == 2) ? -m : 0.f;
            const float il = (ql == 1) ? -m : (ql == 3) ? m : 0.f;
            const float rh = (qh == 0) ? m : (qh == 2) ? -m : 0.f;
            const float ih = (qh == 1) ? -m : (qh == 3) ? m : 0.f;
            Br[k][jj] = hi ? rh : rl;
            Bi[k][jj] = hi ? ih : il;
        }
    }

    // ---- GEMM1: v1 = G0 * v0 (complex, 16 WMMAs; pairs share A operand) ----
    v8f dr = {0.f,0.f,0.f,0.f,0.f,0.f,0.f,0.f};
    v8f di = {0.f,0.f,0.f,0.f,0.f,0.f,0.f,0.f};
    #pragma unroll
    for (int k = 0; k < 4; ++k) {
        dr = wmma4(Ar[k], Br[k], dr);
        di = wmma4(Ar[k], Bi[k], di);
    }
    #pragma unroll
    for (int k = 0; k < 4; ++k) {
        dr = wmma4(An[k], Bi[k], dr);
        di = wmma4(Ai[k], Br[k], di);
    }

    // ---- per-sample RX layer in C/D layout (RX gates commute: do q1,q2,q3
    //      in-lane first, then fuse q0 (cross-half) into the D->B relayout) ----
    float R[8], I[8];
    #pragma unroll
    for (int r = 0; r < 8; ++r) { R[r] = dr[r]; I[r] = di[r]; }

    // qubit 1: in-lane pairs (r, r+4)
    #pragma unroll
    for (int r = 0; r < 4; ++r) {
        float arr = R[r], aii = I[r], brr = R[r + 4], bii = I[r + 4];
        R[r]     = c1 * arr + s1 * bii;  I[r]     = c1 * aii - s1 * brr;
        R[r + 4] = c1 * brr + s1 * aii;  I[r + 4] = c1 * bii - s1 * arr;
    }
    // qubit 2: pairs (r, r+2), r in {0,1,4,5}
    #pragma unroll
    for (int t2 = 0; t2 < 4; ++t2) {
        int r = (t2 & 1) + (t2 >> 1) * 4;
        float arr = R[r], aii = I[r], brr = R[r + 2], bii = I[r + 2];
        R[r]     = c2 * arr + s2 * bii;  I[r]     = c2 * aii - s2 * brr;
        R[r + 2] = c2 * brr + s2 * aii;  I[r + 2] = c2 * bii - s2 * arr;
    }
    // qubit 3: pairs (r, r+1), r in {0,2,4,6}
    #pragma unroll
    for (int t3 = 0; t3 < 4; ++t3) {
        int r = t3 * 2;
        float arr = R[r], aii = I[r], brr = R[r + 1], bii = I[r + 1];
        R[r]     = c3 * arr + s3 * bii;  I[r]     = c3 * aii - s3 * brr;
        R[r + 1] = c3 * brr + s3 * aii;  I[r + 1] = c3 * bii - s3 * arr;
    }

    // single half-swap of pre-q0 values
    float SR[8], SI[8];
    #pragma unroll
    for (int r = 0; r < 8; ++r) { SR[r] = swap16(R[r]); SI[r] = swap16(I[r]); }

    // fused qubit-0 RX: N = new own-comp values, P = new partner-comp values
    float NR[8], NI[8], PR[8], PI[8];
    #pragma unroll
    for (int r = 0; r < 8; ++r) {
        NR[r] = c0 * R[r]  + s0 * SI[r];
        NI[r] = c0 * I[r]  - s0 * SR[r];
        PR[r] = c0 * SR[r] + s0 * I[r];
        PI[r] = c0 * SI[r] - s0 * R[r];
    }

    // D -> B relayout (chunk k, elem jj needs comp 4k+jj+2h)
    #pragma unroll
    for (int jj = 0; jj < 2; ++jj) {
        Br[0][jj] = hi ? PR[jj + 2] : NR[jj];      Bi[0][jj] = hi ? PI[jj + 2] : NI[jj];
        Br[1][jj] = hi ? PR[jj + 6] : NR[jj + 4];  Bi[1][jj] = hi ? PI[jj + 6] : NI[jj + 4];
        Br[2][jj] = hi ? NR[jj + 2] : PR[jj];      Bi[2][jj] = hi ? NI[jj + 2] : PI[jj];
        Br[3][jj] = hi ? NR[jj + 6] : PR[jj + 4];  Bi[3][jj] = hi ? NI[jj + 6] : PI[jj + 4];
    }

    // ---- A operands for G1 ----
    #pragma unroll
    for (int k = 0; k < 4; ++k) {
        int off = n * 16 + 4 * k + cb;
        Ar[k] = *reinterpret_cast<const v2f*>(g + 3 * 256 + off);
        Ai[k] = *reinterpret_cast<const v2f*>(g + 4 * 256 + off);
        An[k] = *reinterpret_cast<const v2f*>(g + 5 * 256 + off);
    }

    // ---- GEMM2: v3 = G1 * v2 (complex, 16 WMMAs; pairs share A operand) ----
    v8f er = {0.f,0.f,0.f,0.f,0.f,0.f,0.f,0.f};
    v8f ei = {0.f,0.f,0.f,0.f,0.f,0.f,0.f,0.f};
    #pragma unroll
    for (int k = 0; k < 4; ++k) {
        er = wmma4(Ar[k], Br[k], er);
        ei = wmma4(Ar[k], Bi[k], ei);
    }
    #pragma unroll
    for (int k = 0; k < 4; ++k) {
        er = wmma4(An[k], Bi[k], er);
        ei = wmma4(Ai[k], Br[k], ei);
    }

    // ---- probabilities and <Z_i> (butterfly reduction) ----
    float p[8];
    #pragma unroll
    for (int r = 0; r < 8; ++r) p[r] = er[r] * er[r] + ei[r] * ei[r];

    const float t0 = p[0] + p[1], t1 = p[2] + p[3];
    const float t2 = p[4] + p[5], t3 = p[6] + p[7];
    const float u01 = t0 + t1, u23 = t2 + t3;
    float z0 = u01 + u23;                 // qubit 0: sign applied per half below
    float z1 = u01 - u23;                 // qubit 1: +{0..3} -{4..7}
    float z2 = (t0 + t2) - (t1 + t3);     // qubit 2: +{0,1,4,5} -{2,3,6,7}
    const float w0 = p[0] - p[1], w1 = p[2] - p[3];
    const float w2 = p[4] - p[5], w3 = p[6] - p[7];
    float z3 = (w0 + w1) + (w2 + w3);     // qubit 3: +even -odd
    if (hi) z0 = -z0;                     // own components have bit3 == hi
    z0 += swap16(z0);
    z1 += swap16(z1);
    z2 += swap16(z2);
    z3 += swap16(z3);

    if (!hi) {
        float4 o; o.x = z0; o.y = z1; o.z = z2; o.w = z3;
        reinterpret_cast<float4*>(out)[samp] = o;
    }
}

// ---------------------------------------------------------------------------
extern "C" void kernel_launch(void* const* d_in, const int* in_sizes, int n_in,
                              void* d_out, int out_size, void* d_ws, size_t ws_size,
                              hipStream_t stream) {
    const float* x = (const float*)d_in[0];   // [B,4] float32
    const float* w = (const float*)d_in[1];   // [2,4,2] float32
    float* g   = (float*)d_ws;                // 6*256 floats of unitary tables
    float* out = (float*)d_out;               // [B,4] float32

    int B = in_sizes[0] / 4;                  // 1048576
    build_G<<<1, 512, 0, stream>>>(w, g);
    qml_kernel<<<B / 128, 256, 0, stream>>>(x, g, out);
}